// CRPS_25177098289693
// MI455X (gfx1250) — compile-verified
//
#include <hip/hip_runtime.h>
#include <cstdint>

#define NB 8
#define H 1024
#define Wd 1024
#define N_ELEM (H * Wd)
#define NUMBER 1000
#define TW 64
#define TH 64
#define IN_ROWS 72      // TH + 8 halo rows
#define IN_STRIDE 104   // 96 used cols (TW + 32 alignment margin), padded, mult of 4
#define HS_ROWS 72
#define HS_STRIDE 68    // mult of 4 (16B-aligned rows), 68 mod 64 = 4 bank rotation
#define NTHREADS 256

// ---- ordered-float <-> uint encoding for atomic min/max -------------------
__device__ __forceinline__ unsigned enc_f(float f) {
    unsigned u = __float_as_uint(f);
    return (u & 0x80000000u) ? ~u : (u | 0x80000000u);
}
__device__ __forceinline__ float dec_f(unsigned e) {
    unsigned u = (e & 0x80000000u) ? (e & 0x7FFFFFFFu) : ~e;
    return __uint_as_float(u);
}

// ---- async-stage one 72x96 input tile (with zeroed out-of-image halo) -----
__device__ __forceinline__ void load_tile_async(const float* __restrict__ img,
                                                int i0, int j0,
                                                float* lds_in, int tid) {
    // zero-fill (halo outside the image must contribute 0 to sums)
    float4 z4 = make_float4(0.f, 0.f, 0.f, 0.f);
    for (int k = tid; k < (IN_ROWS * IN_STRIDE) / 4; k += NTHREADS)
        reinterpret_cast<float4*>(lds_in)[k] = z4;
    __syncthreads();

    int rlo = max(0, i0 - 4), rhi = min(H, i0 + TH + 4);
    int clo = max(0, j0 - 16), chi = min(Wd, j0 + TW + 16);  // 16-float aligned span
    int nrows = rhi - rlo;
    int nch = (chi - clo) >> 2;  // b128 chunks per row
    int total = nrows * nch;
    unsigned ldsbase = (unsigned)(size_t)lds_in;  // low 32 bits of flat = LDS offset

    for (int k = tid; k < total; k += NTHREADS) {
        int r = k / nch;
        int c = k - r * nch;
        int grow = rlo + r;
        int gcol = clo + (c << 2);
        unsigned loff = ldsbase +
            (unsigned)(((grow - (i0 - 4)) * IN_STRIDE + (gcol - (j0 - 16))) * 4);
        unsigned goff = (unsigned)((grow * Wd + gcol) * 4);
        // CDNA5 async global->LDS copy (GVS mode: saddr pair + 32-bit voffset)
        asm volatile("global_load_async_to_lds_b128 %0, %1, %2"
                     :: "v"(loff), "v"(goff), "s"(img) : "memory");
    }
    asm volatile("s_wait_asynccnt 0x0" ::: "memory");
    __syncthreads();
}

// ---- horizontal 9-tap sums, vectorized: 3x ds_load_b128 -> 4 outputs ------
__device__ __forceinline__ void hsum_tile(const float* lds_in, float* lds_hs, int tid) {
    // 72 rows * 16 quads = 1152 work items
    for (int k = tid; k < HS_ROWS * (TW / 4); k += NTHREADS) {
        int r = k >> 4;
        int q = (k & 15) << 2;  // output column base (mult of 4)
        // outputs q..q+3 need lds cols q+12 .. q+23 (12 floats, 16B aligned)
        const float4* p = reinterpret_cast<const float4*>(
            &lds_in[r * IN_STRIDE + q + 12]);
        float4 a = p[0], b = p[1], c = p[2];
        float s0 = a.x + a.y + a.z + a.w + b.x + b.y + b.z + b.w + c.x;
        float s1 = s0 - a.x + c.y;
        float s2 = s1 - a.y + c.z;
        float s3 = s2 - a.z + c.w;
        *reinterpret_cast<float4*>(&lds_hs[r * HS_STRIDE + q]) =
            make_float4(s0, s1, s2, s3);
    }
    __syncthreads();
}

// ---- pass 1: dist = v - avgpool(v), per-batch global min/max --------------
__global__ __launch_bounds__(NTHREADS)
void kminmax(const float* __restrict__ pred, const float* __restrict__ targ,
             unsigned* __restrict__ mm) {
    __shared__ __align__(16) float lds_in[IN_ROWS * IN_STRIDE];
    __shared__ __align__(16) float lds_hs[HS_ROWS * HS_STRIDE];
    __shared__ float redmin[NTHREADS], redmax[NTHREADS];
    int tid = threadIdx.x;
    int z = blockIdx.z, t = z >> 3, b = z & 7;
    const float* img = (t ? targ : pred) + (size_t)b * N_ELEM;
    int i0 = blockIdx.y * TH, j0 = blockIdx.x * TW;

    load_tile_async(img, i0, j0, lds_in, tid);
    hsum_tile(lds_in, lds_hs, tid);

    int col = tid & 63, rbase = (tid >> 6) * 16;
    float lmin = 3.4e38f, lmax = -3.4e38f;
    float vs = 0.f;
#pragma unroll
    for (int d = 0; d < 9; ++d) vs += lds_hs[(rbase + d) * HS_STRIDE + col];
#pragma unroll
    for (int rr = 0; rr < 16; ++rr) {
        int ti = rbase + rr;
        int gi = i0 + ti, gj = j0 + col;
        float v = lds_in[(ti + 4) * IN_STRIDE + col + 16];
        int rh = min(gi + 4, H - 1) - max(gi - 4, 0) + 1;
        int rw = min(gj + 4, Wd - 1) - max(gj - 4, 0) + 1;
        float dist = v - vs / (float)(rh * rw);
        lmin = fminf(lmin, dist);
        lmax = fmaxf(lmax, dist);
        if (rr < 15)
            vs += lds_hs[(ti + 9) * HS_STRIDE + col] - lds_hs[ti * HS_STRIDE + col];
    }
    redmin[tid] = lmin; redmax[tid] = lmax;
    __syncthreads();
    for (int s = NTHREADS / 2; s > 0; s >>= 1) {
        if (tid < s) {
            redmin[tid] = fminf(redmin[tid], redmin[tid + s]);
            redmax[tid] = fmaxf(redmax[tid], redmax[tid + s]);
        }
        __syncthreads();
    }
    if (tid == 0) {
        atomicMin(&mm[2 * b], enc_f(redmin[0]));
        atomicMax(&mm[2 * b + 1], enc_f(redmax[0]));
    }
}

// ---- pass 2: recompute dist, bin into per-batch 1000-bin histogram --------
__global__ __launch_bounds__(NTHREADS)
void khist(const float* __restrict__ pred, const float* __restrict__ targ,
           const unsigned* __restrict__ mm, unsigned* __restrict__ hist) {
    __shared__ __align__(16) float lds_in[IN_ROWS * IN_STRIDE];
    __shared__ __align__(16) float lds_hs[HS_ROWS * HS_STRIDE];
    __shared__ unsigned lhist[NUMBER];
    int tid = threadIdx.x;
    int z = blockIdx.z, t = z >> 3, b = z & 7;
    const float* img = (t ? targ : pred) + (size_t)b * N_ELEM;
    unsigned* gh = hist + (size_t)t * (NB * NUMBER) + b * NUMBER;
    float xmin = dec_f(mm[2 * b]);
    float xmax = dec_f(mm[2 * b + 1]);
    float scale = (xmax > xmin) ? (float)(NUMBER - 1) / (xmax - xmin) : 0.f;
    for (int k = tid; k < NUMBER; k += NTHREADS) lhist[k] = 0u;
    int i0 = blockIdx.y * TH, j0 = blockIdx.x * TW;

    load_tile_async(img, i0, j0, lds_in, tid);  // barriers also cover lhist init
    hsum_tile(lds_in, lds_hs, tid);

    int col = tid & 63, rbase = (tid >> 6) * 16;
    float vs = 0.f;
#pragma unroll
    for (int d = 0; d < 9; ++d) vs += lds_hs[(rbase + d) * HS_STRIDE + col];
#pragma unroll
    for (int rr = 0; rr < 16; ++rr) {
        int ti = rbase + rr;
        int gi = i0 + ti, gj = j0 + col;
        float v = lds_in[(ti + 4) * IN_STRIDE + col + 16];
        int rh = min(gi + 4, H - 1) - max(gi - 4, 0) + 1;
        int rw = min(gj + 4, Wd - 1) - max(gj - 4, 0) + 1;
        float dist = v - vs / (float)(rh * rw);
        // smallest k with x_k >= dist  (searchsorted side='right' equivalence)
        int k0 = (int)ceilf((dist - xmin) * scale);
        k0 = max(0, min(NUMBER - 1, k0));
        atomicAdd(&lhist[k0], 1u);
        if (rr < 15)
            vs += lds_hs[(ti + 9) * HS_STRIDE + col] - lds_hs[ti * HS_STRIDE + col];
    }
    __syncthreads();
    for (int k = tid; k < NUMBER; k += NTHREADS) {
        unsigned c = lhist[k];
        if (c) atomicAdd(&gh[k], c);
    }
}

// ---- pass 3: per-batch scan -> CDFs -> trapezoid -> mean ------------------
__global__ __launch_bounds__(1024)
void kfinal(const unsigned* __restrict__ mm, const unsigned* __restrict__ hist,
            float* __restrict__ out) {
    __shared__ unsigned sb[1024];
    __shared__ float fred[1024];
    __shared__ float dEnd[2];
    int tid = threadIdx.x;
    int b = blockIdx.x;
    const unsigned* hp = hist + b * NUMBER;
    const unsigned* ht = hist + NB * NUMBER + b * NUMBER;

    // inclusive scan of pred histogram
    sb[tid] = (tid < NUMBER) ? hp[tid] : 0u;
    __syncthreads();
    for (int off = 1; off < 1024; off <<= 1) {
        unsigned a = (tid >= off) ? sb[tid - off] : 0u;
        __syncthreads();
        sb[tid] += a;
        __syncthreads();
    }
    unsigned P = sb[tid];
    __syncthreads();
    // inclusive scan of targ histogram
    sb[tid] = (tid < NUMBER) ? ht[tid] : 0u;
    __syncthreads();
    for (int off = 1; off < 1024; off <<= 1) {
        unsigned a = (tid >= off) ? sb[tid - off] : 0u;
        __syncthreads();
        sb[tid] += a;
        __syncthreads();
    }
    unsigned T = sb[tid];

    float d = 0.f;
    if (tid < NUMBER) {
        float c = ((float)(int)P - (float)(int)T) * (1.0f / (float)N_ELEM);
        d = c * c;
    }
    if (tid == 0) dEnd[0] = d;
    if (tid == NUMBER - 1) dEnd[1] = d;
    fred[tid] = d;
    __syncthreads();
    for (int s = 512; s > 0; s >>= 1) {
        if (tid < s) fred[tid] += fred[tid + s];
        __syncthreads();
    }
    if (tid == 0) {
        float xmin = dec_f(mm[2 * b]), xmax = dec_f(mm[2 * b + 1]);
        float delta = (xmax - xmin) * (1.0f / (float)(NUMBER - 1));
        float crps = delta * (fred[0] - 0.5f * (dEnd[0] + dEnd[1]));
        atomicAdd(out, crps * (1.0f / (float)NB));
    }
}

// ---- init: zero out/hist, seed min/max ------------------------------------
__global__ void kinit(unsigned* __restrict__ ws_u32, float* __restrict__ out) {
    int tid = blockIdx.x * blockDim.x + threadIdx.x;
    if (tid == 0) out[0] = 0.f;
    if (tid < 2 * NB) ws_u32[tid] = (tid & 1) ? 0u : 0xFFFFFFFFu;  // min=UMAX, max=0
    for (int k = tid; k < 2 * NB * NUMBER; k += gridDim.x * blockDim.x)
        ws_u32[16 + k] = 0u;
}

extern "C" void kernel_launch(void* const* d_in, const int* in_sizes, int n_in,
                              void* d_out, int out_size, void* d_ws, size_t ws_size,
                              hipStream_t stream) {
    (void)in_sizes; (void)n_in; (void)out_size; (void)ws_size;
    const float* pred = (const float*)d_in[0];
    const float* targ = (const float*)d_in[1];
    float* out = (float*)d_out;
    unsigned* ws = (unsigned*)d_ws;
    unsigned* mm = ws;             // 16 u32: per-batch (min_enc, max_enc)
    unsigned* hist = ws + 16;      // 2 * 8 * 1000 u32

    kinit<<<64, 256, 0, stream>>>(ws, out);
    dim3 grid(Wd / TW, H / TH, 2 * NB);
    kminmax<<<grid, NTHREADS, 0, stream>>>(pred, targ, mm);
    khist<<<grid, NTHREADS, 0, stream>>>(pred, targ, mm, hist);
    kfinal<<<NB, 1024, 0, stream>>>(mm, hist, out);
}